// NeuralNetwork_74466142978489
// MI455X (gfx1250) — compile-verified
//
#include <hip/hip_runtime.h>
#include <hip/hip_bf16.h>
#include <math.h>

// out[B,16] = act_col( A[B,64] @ W[0:64, 496:512] + bias[496:512] )
// Memory-bound (~21 MB @ 23.3 TB/s ~ 0.9us). fp32 WMMA 16x16x4 for the MACs.
// Branchless epilogue, one v_exp + one v_rcp per output element.

typedef __attribute__((ext_vector_type(2))) float v2f;
typedef __attribute__((ext_vector_type(8))) float v8f;

#define N_TOT 512
#define N_INP 64
#define N_OUT 16
#define ROWS_PER_BLOCK 128   // 8 waves * 16 rows

__global__ __launch_bounds__(256) void nn_out_wmma_kernel(
    const float* __restrict__ A,     // [B, 64]
    const float* __restrict__ W,     // [512, 512]
    const float* __restrict__ bias,  // [512]
    const int*   __restrict__ acts,  // [512]
    float* __restrict__ out,         // [B, 16]
    int B)
{
    __shared__ float w_lds[N_INP][N_OUT]; // 4 KB weight slice W[k][496+c]
    __shared__ float b_lds[N_OUT];
    __shared__ int   a_lds[N_OUT];

    const int tid  = threadIdx.x;
    const int wave = tid >> 5;
    const int lane = tid & 31;
    const int col  = lane & 15;            // output column handled by this lane
    const int koff = (lane >> 4) << 1;     // 0 for lanes 0-15, 2 for lanes 16-31
    const int r0   = blockIdx.x * ROWS_PER_BLOCK + wave * 16;

    // Prefetch this wave's A tile (16 rows x 256B) while weights stage to LDS.
    if (r0 + 16 <= B) {
        const float* p = A + (size_t)(r0 + col) * N_INP;
        __builtin_prefetch(p, 0, 0);
        __builtin_prefetch(p + 32, 0, 0);
    }

    // Cooperative stage of W[0:64, 496:512]: 1024 floats, 256 threads x float4.
    {
        const int k  = tid >> 2;
        const int c4 = (tid & 3) << 2;
        const float4 v = *reinterpret_cast<const float4*>(
            W + (size_t)k * N_TOT + (N_TOT - N_OUT) + c4);
        w_lds[k][c4 + 0] = v.x;
        w_lds[k][c4 + 1] = v.y;
        w_lds[k][c4 + 2] = v.z;
        w_lds[k][c4 + 3] = v.w;
    }
    if (tid < N_OUT) {
        b_lds[tid] = bias[(N_TOT - N_OUT) + tid];
        a_lds[tid] = acts[(N_TOT - N_OUT) + tid];
    }
    __syncthreads();

    if (r0 + 16 > B) return;  // wave-uniform: whole wave exits together

    // B-fragments (4x16 K-slices) from LDS: lane<16 holds B[k0..k1][col],
    // lane>=16 holds B[k2..k3][col]. Lowers to ds_load_2addr_b32 clauses.
    v2f bfrag[16];
#pragma unroll
    for (int t = 0; t < 16; ++t) {
        const int k = 4 * t + koff;
        v2f b; b.x = w_lds[k][col]; b.y = w_lds[k + 1][col];
        bfrag[t] = b;
    }

    // A-fragments: issue all 16 NT b64 loads as one clause into distinct
    // registers so the WMMA chain pipelines against loadcnt (no per-iteration
    // full-latency stall). lane<16 -> A[r0+col][4t..4t+1], lane>=16 -> +2.
    const float* arow = A + (size_t)(r0 + col) * N_INP + koff;
    v2f afrag[16];
#pragma unroll
    for (int t = 0; t < 16; ++t)
        afrag[t] =
            __builtin_nontemporal_load(reinterpret_cast<const v2f*>(arow + 4 * t));

    // Accumulator seeded with bias (every row-slot of this lane is column `col`).
    const float bcol = b_lds[col];
    v8f c;
#pragma unroll
    for (int v = 0; v < 8; ++v) c[v] = bcol;

    // K loop: 16 x V_WMMA_F32_16X16X4_F32, back-to-back on the XDL pipe.
#pragma unroll
    for (int t = 0; t < 16; ++t)
        c = __builtin_amdgcn_wmma_f32_16x16x4_f32(
            /*neg_a=*/false, afrag[t], /*neg_b=*/false, bfrag[t],
            /*c_mod=*/(short)0, c, /*reuse_a=*/false, /*reuse_b=*/false);

    // Branchless epilogue. tanh and sigmoid share f(e)=1/(1+e):
    //   tanh(t)    = 1 - 2*f(exp(2t))
    //   sigmoid(t) = f(exp(-t))
    // -> one v_exp_f32 + one v_rcp_f32 per element, selected by lane-uniform
    // predicates; final pick via v_cndmask chain (lanes can never reconverge).
    const int aid = a_lds[col];
    const bool is_tanh  = (aid == 0);
    const bool is_sig   = (aid == 1);
    const bool is_relu  = (aid == 2);
    const bool is_lrelu = (aid == 3);
    const int mo = (lane >> 4) << 3;   // +8 row offset for hi half-lanes
#pragma unroll
    for (int v = 0; v < 8; ++v) {
        const float t   = c[v];
        const float arg = is_tanh ? (2.0f * t) : (-t);
        const float e   = __expf(arg);
        const float fe  = __builtin_amdgcn_rcpf(1.0f + e); // 1/(1+e)
        const float ts  = is_tanh ? __builtin_fmaf(-2.0f, fe, 1.0f) : fe;
        float r = t;                                       // 4: identity
        r = is_lrelu ? ((t > 0.0f) ? t : 0.01f * t) : r;
        r = is_relu  ? fmaxf(t, 0.0f) : r;
        r = (is_tanh || is_sig) ? ts : r;
        __builtin_nontemporal_store(r, &out[(size_t)(r0 + v + mo) * N_OUT + col]);
    }
}

extern "C" void kernel_launch(void* const* d_in, const int* in_sizes, int n_in,
                              void* d_out, int out_size, void* d_ws, size_t ws_size,
                              hipStream_t stream) {
    const float* A    = (const float*)d_in[0];   // input_values [B,64]
    const float* W    = (const float*)d_in[1];   // weight_matrix [512,512]
    const float* bias = (const float*)d_in[2];   // biases [512]
    const int*   acts = (const int*)d_in[3];     // act_ids [512]
    float* out = (float*)d_out;                  // [B,16]

    const int B = in_sizes[0] / N_INP;
    const int blocks = (B + ROWS_PER_BLOCK - 1) / ROWS_PER_BLOCK;
    nn_out_wmma_kernel<<<blocks, 256, 0, stream>>>(A, W, bias, acts, out, B);
}